// ADC_Module_12163347382417
// MI455X (gfx1250) — compile-verified
//
#include <hip/hip_runtime.h>

#define H 256
#define W 256
#define TW 64
#define TH 64
#define NH (TH + 2)      // 66 halo rows
#define LSTRIDE 68       // floats per halo row in LDS (halo col c stored at index c+3)

__device__ __forceinline__ float AD(float p, float q) { return fabsf(p - q); }

__global__ __launch_bounds__(256) void adc_stencil_kernel(const float* __restrict__ x,
                                                          const float* __restrict__ alpha,
                                                          float* __restrict__ out) {
  __shared__ __align__(16) float tile[NH * LSTRIDE];   // 66*68*4 = 17952 B

  const int tid   = threadIdx.x;
  const int plane = blockIdx.z;                // b*64 + c
  const int x0    = blockIdx.x * TW;
  const int y0    = blockIdx.y * TH;

  const float* __restrict__ xp = x + (size_t)plane * (H * W);
  const unsigned long long gbase = (unsigned long long)(uintptr_t)xp;

  // ================= async global->LDS halo fill (ASYNCcnt path) =================
  const int cc = tid & 15;     // vec4 column 0..15 -> halo cols 1+4cc .. 4+4cc
  const int rr = tid >> 4;     // 0..15

  // Region A: interior halo rows 1..64 (gy always in range), 4x b128 per thread.
#pragma unroll
  for (int j = 0; j < 4; ++j) {
    const int r = 1 + rr + 16 * j;
    const unsigned ldsoff = (unsigned)(uintptr_t)(&tile[r * LSTRIDE + 4 + 4 * cc]);
    const unsigned long long ga =
        gbase + (unsigned long long)(((unsigned)((y0 + r - 1) * W + x0 + 4 * cc)) * 4u);
    asm volatile("global_load_async_to_lds_b128 %0, %1, off"
                 :: "v"(ldsoff), "v"(ga) : "memory");
  }

  // Region B: top (r=0) / bottom (r=65) halo rows, cols 1..64, b128 for 32 lanes.
  if (tid < 32) {
    const int r  = (tid < 16) ? 0 : (TH + 1);
    const int gy = y0 + r - 1;
    float* lp = &tile[r * LSTRIDE + 4 + 4 * cc];
    if ((unsigned)gy < (unsigned)H) {
      const unsigned long long ga =
          gbase + (unsigned long long)(((unsigned)(gy * W + x0 + 4 * cc)) * 4u);
      asm volatile("global_load_async_to_lds_b128 %0, %1, off"
                   :: "v"((unsigned)(uintptr_t)lp), "v"(ga) : "memory");
    } else {
      *(float4*)lp = float4{0.f, 0.f, 0.f, 0.f};  // zero padding row
    }
  }

  // Region C: left (c=0 -> idx 3) / right (c=65 -> idx 68) halo cols, rows 0..65.
  if (tid < 2 * NH) {
    const int side = (tid >= NH) ? 1 : 0;
    const int r    = tid - side * NH;            // 0..65
    const int gy   = y0 + r - 1;
    const int gx   = side ? (x0 + TW) : (x0 - 1);
    const int lidx = r * LSTRIDE + (side ? (4 + TW) : 3);
    if (((unsigned)gy < (unsigned)H) & ((unsigned)gx < (unsigned)W)) {
      const unsigned long long ga =
          gbase + (unsigned long long)(((unsigned)(gy * W + gx)) * 4u);
      asm volatile("global_load_async_to_lds_b32 %0, %1, off"
                   :: "v"((unsigned)(uintptr_t)&tile[lidx]), "v"(ga) : "memory");
    } else {
      tile[lidx] = 0.0f;  // zero padding corners/edges
    }
  }

#if __has_builtin(__builtin_amdgcn_s_wait_asynccnt)
  __builtin_amdgcn_s_wait_asynccnt(0);
#else
  asm volatile("s_wait_asynccnt 0" ::: "memory");
#endif
  __syncthreads();

  // ================= compute: each thread -> 4 rows x 4 cols, shared diffs =======
  const int tx = tid & 15;   // output cols 4tx..4tx+3
  const int ty = tid >> 4;   // output rows 4ty..4ty+3
  const float a = alpha[plane & 63];

  const int cbase = 4 * tx + 3;   // LDS col index of leftmost neighbor
  const float* lrow = &tile[4 * ty * LSTRIDE + cbase];

  float va[6], vb[6], vc[6];      // 3-row sliding window of halo values
#pragma unroll
  for (int j = 0; j < 6; ++j) {
    va[j] = lrow[j];
    vb[j] = lrow[LSTRIDE + j];
    vc[j] = lrow[2 * LSTRIDE + j];
  }

  // Carried "up" diffs relative to current output row R:
  //   vtp[j] = |v[R+1][j+1]-v[R][j+1]|, d1p[j] = |v[R+1][j+1]-v[R][j]| (\),
  //   d2p[j] = |v[R+1][j]-v[R][j+1]| (/)
  float vtp[4], d1p[5], d2p[5];
#pragma unroll
  for (int j = 0; j < 4; ++j) vtp[j] = AD(vb[j + 1], va[j + 1]);
#pragma unroll
  for (int j = 0; j < 5; ++j) { d1p[j] = AD(vb[j + 1], va[j]); d2p[j] = AD(vb[j], va[j + 1]); }

  float* dst = out + (size_t)plane * (H * W) + (size_t)(y0 + 4 * ty) * W + (x0 + 4 * tx);

#pragma unroll
  for (int q = 0; q < 4; ++q) {
    // New diffs for this output row: middle row vb, bottom row vc.
    float hh[5], vtn[4], d1n[5], d2n[5];
#pragma unroll
    for (int j = 0; j < 5; ++j) hh[j] = AD(vb[j + 1], vb[j]);        // horizontal
#pragma unroll
    for (int j = 0; j < 4; ++j) vtn[j] = AD(vc[j + 1], vb[j + 1]);   // vertical down
#pragma unroll
    for (int j = 0; j < 5; ++j) { d1n[j] = AD(vc[j + 1], vb[j]); d2n[j] = AD(vc[j], vb[j + 1]); }

    float4 o;
    float* op = (float*)&o;
#pragma unroll
    for (int k = 0; k < 4; ++k) {
      const float ctr = vb[k + 1];
      const float s = ((hh[k] + hh[k + 1]) + (vtp[k] + vtn[k])) +
                      ((d1p[k] + d1n[k + 1]) + (d2p[k + 1] + d2n[k]));
      op[k] = fmaf(a, s, ctr);
    }
    *(float4*)dst = o;          // coalesced global_store_b128
    dst += W;

    if (q < 3) {                // slide window: carry diffs, load next halo row
#pragma unroll
      for (int j = 0; j < 4; ++j) vtp[j] = vtn[j];
#pragma unroll
      for (int j = 0; j < 5; ++j) { d1p[j] = d1n[j]; d2p[j] = d2n[j]; }
#pragma unroll
      for (int j = 0; j < 6; ++j) {
        va[j] = vb[j];
        vb[j] = vc[j];
        vc[j] = lrow[(q + 3) * LSTRIDE + j];
      }
    }
  }
}

extern "C" void kernel_launch(void* const* d_in, const int* in_sizes, int n_in,
                              void* d_out, int out_size, void* d_ws, size_t ws_size,
                              hipStream_t stream) {
  const float* x     = (const float*)d_in[0];
  const float* alpha = (const float*)d_in[1];
  float* out         = (float*)d_out;

  dim3 grid(W / TW, H / TH, 8 * 64);   // 4 x 4 x 512 = 8192 blocks
  adc_stencil_kernel<<<grid, 256, 0, stream>>>(x, alpha, out);
}